// S2VEncoder_46875273068981
// MI455X (gfx1250) — compile-verified
//
#include <hip/hip_runtime.h>

typedef __attribute__((ext_vector_type(2))) float v2f;
typedef __attribute__((ext_vector_type(8))) float v8f;

#define Hdim 128
#define LDSTRIDE 132   // 128 + 4 pad: rows land on distinct LDS banks

// ---------------- utility kernels ----------------

__global__ void zero_f32(float* __restrict__ p, long n) {
    long i = (long)blockIdx.x * blockDim.x + threadIdx.x;
    long stride = (long)gridDim.x * blockDim.x;
    for (; i < n; i += stride) p[i] = 0.0f;
}

// v[i][h] = sum_k W3[i][h][k] * relu(W4[i][k])   (edge_attr == ones collapses
// the whole edge-embedding pipeline to deg[n] * v[i])
__global__ void compute_v(const float* __restrict__ W3,
                          const float* __restrict__ W4,
                          float* __restrict__ v) {
    int i = blockIdx.x;       // hop
    int h = threadIdx.x;      // output channel
    const float* w3 = W3 + (size_t)i * Hdim * Hdim + (size_t)h * Hdim;
    const float* w4 = W4 + (size_t)i * Hdim;
    float s = 0.0f;
    for (int k = 0; k < Hdim; ++k) s += w3[k] * fmaxf(w4[k], 0.0f);
    v[i * Hdim + h] = s;
}

__global__ void compute_deg(const int* __restrict__ row,
                            float* __restrict__ deg, int E) {
    int e = blockIdx.x * blockDim.x + threadIdx.x;
    if (e < E) unsafeAtomicAdd(&deg[row[e]], 1.0f);
}

// ---------------- SpMM: nbr[row[e]] += h[col[e]] ----------------
// one wave per edge; each lane moves a float4 (128 cols / 32 lanes)
__global__ void spmm_scatter(const float* __restrict__ h,
                             const int* __restrict__ row,
                             const int* __restrict__ col,
                             float* __restrict__ nbr, int E) {
    int e = blockIdx.x * 8 + (threadIdx.x >> 5);
    int lane = threadIdx.x & 31;
    if (e >= E) return;
    int r = row[e];
    int c = col[e];
    const float4 val = *(const float4*)(h + (size_t)c * Hdim + lane * 4);
    float* dst = nbr + (size_t)r * Hdim + lane * 4;
    unsafeAtomicAdd(dst + 0, val.x);
    unsafeAtomicAdd(dst + 1, val.y);
    unsafeAtomicAdd(dst + 2, val.z);
    unsafeAtomicAdd(dst + 3, val.w);
}

// ---------------- fused hop GEMM (WMMA f32 16x16x4) ----------------
// h_out = relu( x @ W1^T + nbr @ W2^T + deg (outer) v )
__global__ void __launch_bounds__(256, 2)
fused_hop(const float* __restrict__ x, const float* __restrict__ nbr,
          const float* __restrict__ W1, const float* __restrict__ W2,
          const float* __restrict__ vvec, const float* __restrict__ deg,
          float* __restrict__ hout, int N, int use_nbr) {
    __shared__ __align__(16) float xs[16 * LDSTRIDE];
    __shared__ __align__(16) float ns[16 * LDSTRIDE];
    __shared__ float degs[16];

    const int t = threadIdx.x;
    const int base = blockIdx.x * 16;

    // cooperative float4 load of the 16x128 A-tiles into padded LDS
    for (int i = t; i < 16 * Hdim / 4; i += 256) {
        int r = i >> 5;               // 32 float4 per row
        int c4 = i & 31;
        int rr = base + r; if (rr >= N) rr = N - 1;
        ((float4*)(xs + r * LDSTRIDE))[c4] =
            ((const float4*)(x + (size_t)rr * Hdim))[c4];
        if (use_nbr)
            ((float4*)(ns + r * LDSTRIDE))[c4] =
                ((const float4*)(nbr + (size_t)rr * Hdim))[c4];
    }
    if (t < 16) {
        int rr = base + t; if (rr >= N) rr = N - 1;
        degs[t] = deg[rr];
    }
    __syncthreads();

    const int wave  = t >> 5;
    const int lane  = t & 31;
    const int mrow  = lane & 15;            // A: M index
    const int khalf = (lane >> 4) << 1;     // A/B: K sub-offset (0 or 2)
    const int coln  = lane & 15;            // B: N index
    const int ncol  = wave << 4;            // this wave's 16-col output tile

    const float* w1c = W1 + (size_t)(ncol + coln) * Hdim;  // B[k][n] = W1[n][k]
    const float* w2c = W2 + (size_t)(ncol + coln) * Hdim;

    v8f acc = {0.f, 0.f, 0.f, 0.f, 0.f, 0.f, 0.f, 0.f};

#pragma unroll
    for (int ks = 0; ks < 32; ++ks) {       // K = 128 = 32 steps of 4
        const int k = (ks << 2) + khalf;
        v2f a, b;
        a.x = xs[mrow * LDSTRIDE + k];
        a.y = xs[mrow * LDSTRIDE + k + 1];
        b.x = w1c[k];
        b.y = w1c[k + 1];
        acc = __builtin_amdgcn_wmma_f32_16x16x4_f32(
            false, a, false, b, (short)0, acc, false, false);
    }
    if (use_nbr) {
#pragma unroll
        for (int ks = 0; ks < 32; ++ks) {
            const int k = (ks << 2) + khalf;
            v2f a, b;
            a.x = ns[mrow * LDSTRIDE + k];
            a.y = ns[mrow * LDSTRIDE + k + 1];
            b.x = w2c[k];
            b.y = w2c[k + 1];
            acc = __builtin_amdgcn_wmma_f32_16x16x4_f32(
                false, a, false, b, (short)0, acc, false, false);
        }
    }

    // D layout: VGPR j -> M = j + 8*(lane/16), N = ncol + lane%16
    const int nOut = ncol + (lane & 15);
    const int mOff = (lane >> 4) << 3;
    const float vn = vvec[nOut];
#pragma unroll
    for (int j = 0; j < 8; ++j) {
        const int m = mOff + j;
        if (base + m < N) {
            float val = acc[j] + degs[m] * vn;
            hout[(size_t)(base + m) * Hdim + nOut] = fmaxf(val, 0.0f);
        }
    }
}

// ---------------- host-side orchestration ----------------

extern "C" void kernel_launch(void* const* d_in, const int* in_sizes, int n_in,
                              void* d_out, int out_size, void* d_ws, size_t ws_size,
                              hipStream_t stream) {
    const float* x   = (const float*)d_in[0];
    const int*  eidx = (const int*)d_in[1];   // [2, E]
    const float* W1  = (const float*)d_in[2]; // [3,128,128]
    const float* W2  = (const float*)d_in[3]; // [3,128,128]
    const float* W3  = (const float*)d_in[4]; // [3,128,128]
    const float* W4  = (const float*)d_in[5]; // [3,128,1]
    float* out = (float*)d_out;

    const int N = in_sizes[0] / Hdim;
    const int E = in_sizes[1] / 2;
    const int* row = eidx;
    const int* col = eidx + E;

    // workspace: h [N*128], nbr [N*128], deg [N], v [3*128]
    float* h   = (float*)d_ws;
    float* nbr = h + (size_t)N * Hdim;
    float* deg = nbr + (size_t)N * Hdim;
    float* v   = deg + N;

    const size_t HF = (size_t)Hdim * Hdim;
    const int gemmGrid = (N + 15) / 16;
    const int spmmGrid = (E + 7) / 8;

    // precompute degrees and the collapsed edge vectors v_i
    zero_f32<<<256, 256, 0, stream>>>(deg, N);
    compute_deg<<<(E + 255) / 256, 256, 0, stream>>>(row, deg, E);
    compute_v<<<3, 128, 0, stream>>>(W3, W4, v);

    // hop 0: h starts at zero -> neighbor term vanishes
    fused_hop<<<gemmGrid, 256, 0, stream>>>(x, nbr, W1, W2, v, deg, h, N, 0);

    // hop 1
    zero_f32<<<2048, 256, 0, stream>>>(nbr, (long)N * Hdim);
    spmm_scatter<<<spmmGrid, 256, 0, stream>>>(h, row, col, nbr, E);
    fused_hop<<<gemmGrid, 256, 0, stream>>>(x, nbr, W1 + HF, W2 + HF,
                                            v + Hdim, deg, h, N, 1);

    // hop 2: write final result straight to d_out
    zero_f32<<<2048, 256, 0, stream>>>(nbr, (long)N * Hdim);
    spmm_scatter<<<spmmGrid, 256, 0, stream>>>(h, row, col, nbr, E);
    fused_hop<<<gemmGrid, 256, 0, stream>>>(x, nbr, W1 + 2 * HF, W2 + 2 * HF,
                                            v + 2 * Hdim, deg, out, N, 1);
}